// gltrivmlp_41936060678627
// MI455X (gfx1250) — compile-verified
//
#include <hip/hip_runtime.h>

// ---------------- CDNA5 / gfx1250 vector types ----------------
typedef float v2f __attribute__((ext_vector_type(2)));
typedef float v4f __attribute__((ext_vector_type(4)));
typedef float v8f __attribute__((ext_vector_type(8)));
typedef int   v4i __attribute__((ext_vector_type(4)));

#define GS          1024
#define PARAM       64
#define NROWS       (32 * 1024)          // B * GS rows
#define TILE_ROWS   16
#define NTILES      (NROWS / TILE_ROWS)  // 2048
#define WAVES_PB    8
#define COLS_PW     (GS / WAVES_PB)      // 128 columns per wave
#define INV_KDIV    (1.0f / 16.0f)

#if __has_builtin(__builtin_amdgcn_global_load_async_to_lds_b128) && \
    __has_builtin(__builtin_amdgcn_s_wait_asynccnt)
#define USE_ASYNC_LDS 1
#else
#define USE_ASYNC_LDS 0
#endif

typedef __attribute__((address_space(1))) v4i* gptr_v4i;
typedef __attribute__((address_space(3))) v4i* lptr_v4i;

__device__ __forceinline__ float clip1(float x) {
    return fminf(fmaxf(x, -1.0f), 1.0f);
}

__global__ __launch_bounds__(256) void gltrivmlp_kernel(
        const float* __restrict__ mat,   // [NROWS][GS]
        const float* __restrict__ val,   // [NROWS][PARAM]
        float* __restrict__ out)         // [NROWS][PARAM]
{
    __shared__ float smat[TILE_ROWS][GS];           // 64 KB staged tile
    __shared__ float spartial[WAVES_PB][TILE_ROWS]; // per-wave partial degs

    const int tid  = threadIdx.x;
    const int wave = tid >> 5;
    const int lane = tid & 31;
    const size_t rowbase = (size_t)blockIdx.x * TILE_ROWS;

    // ---------------- Stage 16x1024 f32 tile into LDS (coalesced) ----------
    // Wave w copies columns [w*128, w*128+128) of every row: 32 lanes x 16B
    // = 512B contiguous per instruction (async DMA into LDS, ASYNCcnt-tracked).
    {
        const int c0 = wave * COLS_PW + lane * 4;
        #pragma unroll
        for (int r = 0; r < TILE_ROWS; ++r) {
            const float* gp = mat + (rowbase + r) * GS + c0;
#if USE_ASYNC_LDS
            __builtin_amdgcn_global_load_async_to_lds_b128(
                (gptr_v4i)gp, (lptr_v4i)&smat[r][c0], 0, 0);
#else
            v4f v = __builtin_nontemporal_load((const v4f*)gp);
            *(v4f*)&smat[r][c0] = v;
#endif
        }
#if USE_ASYNC_LDS
        __builtin_amdgcn_s_wait_asynccnt(0);
#endif
    }
    __syncthreads();

    // ---------------- Row-sum via V_WMMA_F32_16X16X4_F32 -------------------
    // A = 16x4 tile of mat (f32 A-layout: lanes 0-15 = rows, K split across
    // lane halves), B = ones(4x16)  =>  D[m][n] = partial rowsum, replicated
    // over all 16 columns. 4 independent accumulator chains.
    const int mrow  = lane & 15;
    const int khalf = lane >> 4; // 0: K=k0+0..1, 1: K=k0+2..3
    const float* lp = &smat[mrow][wave * COLS_PW + khalf * 2];

    const v2f ones = {1.0f, 1.0f};
    v8f c0 = {}, c1 = {}, c2 = {}, c3 = {};

    #pragma unroll
    for (int k = 0; k < COLS_PW; k += 16) {
        v2f a0 = *(const v2f*)(lp + k + 0);
        v2f a1 = *(const v2f*)(lp + k + 4);
        v2f a2 = *(const v2f*)(lp + k + 8);
        v2f a3 = *(const v2f*)(lp + k + 12);
        c0 = __builtin_amdgcn_wmma_f32_16x16x4_f32(false, a0, false, ones,
                                                   (short)0, c0, false, false);
        c1 = __builtin_amdgcn_wmma_f32_16x16x4_f32(false, a1, false, ones,
                                                   (short)0, c1, false, false);
        c2 = __builtin_amdgcn_wmma_f32_16x16x4_f32(false, a2, false, ones,
                                                   (short)0, c2, false, false);
        c3 = __builtin_amdgcn_wmma_f32_16x16x4_f32(false, a3, false, ones,
                                                   (short)0, c3, false, false);
    }
    v8f acc = (c0 + c1) + (c2 + c3);

    // C/D layout: lane<16 VGPR g holds D[M=g][N=lane]  (= partial deg row g)
    //             lane>=16 VGPR g holds D[M=g+8][...]  (= partial deg row g+8)
    if (lane == 0 || lane == 16) {
        const int base = (lane >> 4) * 8;
        #pragma unroll
        for (int g = 0; g < 8; ++g) spartial[wave][base + g] = acc[g];
    }
    __syncthreads();

    // ---------------- Collapsed iteration (wave 0 only) ---------------------
    // out[r][j]      = con[j]
    // out[r][16+j]   = g(con[j]),  g(x) = clip(x * deg/16, -1, 1)
    // out[r][32+j]   = g^2(con[j]), out[r][48+j] = g^3(con[j])
    // (fixed point of the 10-iteration shift-by-16 recurrence: every column
    //  reaches the frozen con region in <= 3 steps, and 10 >= 3.)
    if (wave == 0) {
        float deg = 0.0f;
        #pragma unroll
        for (int w = 0; w < WAVES_PB; ++w) deg += spartial[w][lane & 15];
        // lane L now holds deg of tile-row (L & 15)

        const int j    = lane & 15;
        const int half = (lane >> 4) * 8; // lanes 16-31 handle rows m+8

        #pragma unroll
        for (int m = 0; m < 8; ++m) {
            const float s = __shfl(deg, m + half, 32) * INV_KDIV;
            const size_t row = rowbase + m + half;
            const float con = val[row * PARAM + j];
            const float x1 = clip1(con * s);
            const float x2 = clip1(x1 * s);
            const float x3 = clip1(x2 * s);
            float* op = out + row * PARAM;
            op[j]      = con;
            op[16 + j] = x1;
            op[32 + j] = x2;
            op[48 + j] = x3;
        }
    }
}

extern "C" void kernel_launch(void* const* d_in, const int* in_sizes, int n_in,
                              void* d_out, int out_size, void* d_ws, size_t ws_size,
                              hipStream_t stream) {
    const float* mat = (const float*)d_in[0]; // [32,1024,1024]
    const float* val = (const float*)d_in[1]; // [32,1024,64]
    float* out = (float*)d_out;               // [32,1024,64]
    (void)in_sizes; (void)n_in; (void)out_size; (void)d_ws; (void)ws_size;

    gltrivmlp_kernel<<<NTILES, 256, 0, stream>>>(mat, val, out);
}